// KalmanNet_58755152609542
// MI455X (gfx1250) — compile-verified
//
#include <hip/hip_runtime.h>
#include <hip/hip_bf16.h>

typedef __attribute__((ext_vector_type(2))) float v2f;
typedef __attribute__((ext_vector_type(8))) float v8f;

#define BATCH 32768
#define DSRC  16
#define TROWS 16
#define LDA   116     // activation leading dim: 116 mod 64 = 52 -> 16 distinct banks, 8B-aligned rows
#define NWAVE 2
#define GRIDY 128     // 2048 tiles / (128*2) = 8 iterations per wave

// ---------------- LDS weight arena (floats), zero-padded [out_pad][in_pad] ----------------
constexpr int OFF_FC1W  = 0;                   // [48][4]
constexpr int OFF_FC1B  = OFF_FC1W  + 192;
constexpr int OFF_FC2W  = OFF_FC1B  + 48;      // [48][4]
constexpr int OFF_FC2B  = OFF_FC2W  + 192;
constexpr int OFF_FC3W  = OFF_FC2B  + 48;      // [16][4]
constexpr int OFF_FC3B  = OFF_FC3W  + 64;
constexpr int OFF_FC4W  = OFF_FC3B  + 16;      // [48][4]
constexpr int OFF_FC4B  = OFF_FC4W  + 192;
constexpr int OFF_FC5AW = OFF_FC4B  + 48;      // [112][12] (ldw 12: conflict-free)
constexpr int OFF_FC5AB = OFF_FC5AW + 1344;
constexpr int OFF_FC5BW = OFF_FC5AB + 112;     // [16][100]
constexpr int OFF_FC5BB = OFF_FC5BW + 1600;
constexpr int OFF_FC6W  = OFF_FC5BB + 16;      // [16][4]
constexpr int OFF_FC6B  = OFF_FC6W  + 64;
constexpr int OFF_FC7W  = OFF_FC6B  + 16;      // [16][12]
constexpr int OFF_FC7B  = OFF_FC7W  + 192;
constexpr int OFF_G1WIH = OFF_FC7B  + 16;      // [16][44] (40 -> ldw 44, conflict-free)
constexpr int OFF_G1WHH = OFF_G1WIH + 704;     // [16][4]
constexpr int OFF_G1BIH = OFF_G1WHH + 64;
constexpr int OFF_G1BHH = OFF_G1BIH + 16;
constexpr int OFF_G2WIH = OFF_G1BHH + 16;      // [16][44]
constexpr int OFF_G2WHH = OFF_G2WIH + 704;
constexpr int OFF_G2BIH = OFF_G2WHH + 64;
constexpr int OFF_G2BHH = OFF_G2BIH + 16;
constexpr int OFF_G3WIH = OFF_G2BHH + 16;      // [16][44] (41 -> 44)
constexpr int OFF_G3WHH = OFF_G3WIH + 704;     // [16][4]
constexpr int OFF_G3BIH = OFF_G3WHH + 64;
constexpr int OFF_G3BHH = OFF_G3BIH + 16;
constexpr int W_TOTAL   = OFF_G3BHH + 16;      // 6560 floats

// per-wave scratch
constexpr int WV_A  = 0;
constexpr int WV_B  = TROWS * LDA;             // 1856
constexpr int WV_HQ = 2 * TROWS * LDA;         // [16][4]
constexpr int WV_HS = WV_HQ + 64;
constexpr int WV_SN = WV_HS + 64;
constexpr int WV_SZ = WV_SN + 64;              // 3904
constexpr int SMEM_FLOATS = W_TOTAL + NWAVE * WV_SZ;   // 14368 floats = 57472 B

struct KNParams {
  const float *del_y_til, *del_y, *del_x_til, *del_x_hat, *Q, *Sigma, *S;
  const float *fc1_w, *fc1_b, *fc2_w, *fc2_b, *fc3_w, *fc3_b, *fc4_w, *fc4_b;
  const float *fc5a_w, *fc5a_b, *fc5b_w, *fc5b_b, *fc6_w, *fc6_b, *fc7_w, *fc7_b;
  const float *g1wih, *g1whh, *g1bih, *g1bhh;
  const float *g2wih, *g2whh, *g2bih, *g2bhh;
  const float *g3wih, *g3whh, *g3bih, *g3bhh;
  float* out;
};

// ---------------- weight preload ----------------
__device__ __forceinline__ void load_w(float* dst, const float* src,
                                       int out_f, int in_f, int out_pad, int in_pad, int tid) {
  for (int e = tid; e < out_pad * in_pad; e += 64) {
    int r = e / in_pad, c = e % in_pad;
    dst[e] = (r < out_f && c < in_f) ? src[r * in_f + c] : 0.f;
  }
}
__device__ __forceinline__ void load_bias(float* dst, const float* src, int nf, int npad, int tid) {
  for (int e = tid; e < npad; e += 64) dst[e] = (e < nf) ? src[e] : 0.f;
}

// ---------------- WMMA tile GEMM (compile-time dims -> unrolled, immediate offsets) ----------
// D[16 rows x NPAD] = A[16 x KPAD] * Wm[NPAD x KPAD]^T  via V_WMMA_F32_16X16X4_F32.
// f32 frag layouts per CDNA5 ISA: A lanes 0-15 -> row=lane K={kb,kb+1}, lanes 16-31 K={kb+2,kb+3};
// B VGPR0/1 -> K=kb(+2)/kb+1(+3), N=lane%16;  D VGPR r -> row r / r+8 per lane half.
template<int KPAD, int NPAD, int LDAA, int LDW, int LDO>
__device__ __forceinline__ void layer_gemm(const float* __restrict__ A,
                                           const float* __restrict__ Wm,
                                           float* __restrict__ O, int lane) {
  const int half = lane >> 4;
  const int l16  = lane & 15;
  const float* arow = A + l16 * LDAA + half * 2;
#pragma unroll
  for (int nb = 0; nb < NPAD; nb += 16) {
    const float* wrow = Wm + (nb + l16) * LDW + half * 2;
    v8f acc = {0.f, 0.f, 0.f, 0.f, 0.f, 0.f, 0.f, 0.f};
#pragma unroll
    for (int kb = 0; kb < KPAD; kb += 4) {
      v2f a, b;
      a.x = arow[kb]; a.y = arow[kb + 1];
      b.x = wrow[kb]; b.y = wrow[kb + 1];
      acc = __builtin_amdgcn_wmma_f32_16x16x4_f32(false, a, false, b, (short)0, acc,
                                                  false, false);
    }
    float* orow = O + half * 8 * LDO + nb + l16;
#pragma unroll
    for (int r = 0; r < 8; ++r) orow[r * LDO] = acc[r];
  }
}

// bias+relu, NF even: lanes split row halves, no div/mod
template<int NF>
__device__ __forceinline__ void bias_relu_half(float* Buf, const float* bias, int lane) {
  static_assert((NF & 1) == 0, "even NF");
  const int r  = lane & 15;
  const int cb = (lane >> 4) * (NF / 2);
#pragma unroll
  for (int k = 0; k < NF / 2; ++k) {
    const int c = cb + k;
    float v = Buf[r * LDA + c] + bias[c];
    Buf[r * LDA + c] = v > 0.f ? v : 0.f;
  }
}

// torch GRU (gate order r,z,n): h' = (1-z)*tanh(i_n + r*h_n) + z*h
__device__ __forceinline__ float gru_elem(float ir, float iz, float in_,
                                          float hr, float hz, float hn, float hprev) {
  float rg = 1.f / (1.f + __expf(-(ir + hr)));
  float zg = 1.f / (1.f + __expf(-(iz + hz)));
  float cand = tanhf(in_ + rg * hn);
  return (1.f - zg) * cand + zg * hprev;
}

template<int H>   // H = 4 or 1; GI/GH have ld = LDA, Hst ld = 4
__device__ __forceinline__ void gru_combine(const float* GI, const float* GH,
                                            const float* bih, const float* bhh,
                                            float* Hst, int lane) {
  if constexpr (H == 4) {
    const int r  = lane & 15;
    const int j0 = (lane >> 4) * 2;
#pragma unroll
    for (int u = 0; u < 2; ++u) {
      const int j = j0 + u;
      float v = gru_elem(GI[r * LDA + j] + bih[j],
                         GI[r * LDA + 4 + j] + bih[4 + j],
                         GI[r * LDA + 8 + j] + bih[8 + j],
                         GH[r * LDA + j] + bhh[j],
                         GH[r * LDA + 4 + j] + bhh[4 + j],
                         GH[r * LDA + 8 + j] + bhh[8 + j],
                         Hst[r * 4 + j]);
      Hst[r * 4 + j] = v;
    }
  } else {
    if (lane < 16) {
      const int r = lane;
      float v = gru_elem(GI[r * LDA + 0] + bih[0],
                         GI[r * LDA + 1] + bih[1],
                         GI[r * LDA + 2] + bih[2],
                         GH[r * LDA + 0] + bhh[0],
                         GH[r * LDA + 1] + bhh[1],
                         GH[r * LDA + 2] + bhh[2],
                         Hst[r * 4 + 0]);
      Hst[r * 4 + 0] = v;
    }
  }
}

// ---------------- kernel ----------------
__global__ __launch_bounds__(NWAVE * 32, 1)
void kalmannet_wmma_kernel(KNParams p) {
  __shared__ float smem[SMEM_FLOATS];
  const int d    = blockIdx.x;
  const int tid  = threadIdx.x;
  const int lane = tid & 31;
  const int wave = tid >> 5;

  float* W  = smem;
  float* wv = smem + W_TOTAL + wave * WV_SZ;
  float* A  = wv + WV_A;     // [16][116]
  float* Bf = wv + WV_B;     // [16][116]
  float* hQ = wv + WV_HQ;    // [16][4]
  float* hS = wv + WV_HS;    // [16][4]
  float* sn = wv + WV_SN;    // [16][4]

  // -------- preload this d's weights (zero-padded) --------
  load_w(W + OFF_FC1W,  p.fc1_w  + d * 80,  40, 2,  48, 4,   tid);
  load_bias(W + OFF_FC1B, p.fc1_b + d * 40, 40, 48, tid);
  load_w(W + OFF_FC2W,  p.fc2_w  + d * 80,  40, 2,  48, 4,   tid);
  load_bias(W + OFF_FC2B, p.fc2_b + d * 40, 40, 48, tid);
  load_w(W + OFF_FC3W,  p.fc3_w  + d * 4,    1, 4,  16, 4,   tid);
  load_bias(W + OFF_FC3B, p.fc3_b + d * 1,   1, 16, tid);
  load_w(W + OFF_FC4W,  p.fc4_w  + d * 80,  40, 2,  48, 4,   tid);
  load_bias(W + OFF_FC4B, p.fc4_b + d * 40, 40, 48, tid);
  load_w(W + OFF_FC5AW, p.fc5a_w + d * 500, 100, 5, 112, 12, tid);
  load_bias(W + OFF_FC5AB, p.fc5a_b + d * 100, 100, 112, tid);
  load_w(W + OFF_FC5BW, p.fc5b_w + d * 200,  2, 100, 16, 100, tid);
  load_bias(W + OFF_FC5BB, p.fc5b_b + d * 2,  2, 16, tid);
  load_w(W + OFF_FC6W,  p.fc6_w  + d * 12,   4, 3,  16, 4,   tid);
  load_bias(W + OFF_FC6B, p.fc6_b + d * 4,   4, 16, tid);
  load_w(W + OFF_FC7W,  p.fc7_w  + d * 32,   4, 8,  16, 12,  tid);
  load_bias(W + OFF_FC7B, p.fc7_b + d * 4,   4, 16, tid);
  load_w(W + OFF_G1WIH, p.g1wih + d * 480, 12, 40, 16, 44, tid);
  load_w(W + OFF_G1WHH, p.g1whh + d * 48,  12, 4,  16, 4,  tid);
  load_bias(W + OFF_G1BIH, p.g1bih + d * 12, 12, 16, tid);
  load_bias(W + OFF_G1BHH, p.g1bhh + d * 12, 12, 16, tid);
  load_w(W + OFF_G2WIH, p.g2wih + d * 528, 12, 44, 16, 44, tid);
  load_w(W + OFF_G2WHH, p.g2whh + d * 48,  12, 4,  16, 4,  tid);
  load_bias(W + OFF_G2BIH, p.g2bih + d * 12, 12, 16, tid);
  load_bias(W + OFF_G2BHH, p.g2bhh + d * 12, 12, 16, tid);
  load_w(W + OFF_G3WIH, p.g3wih + d * 123,  3, 41, 16, 44, tid);
  load_w(W + OFF_G3WHH, p.g3whh + d * 3,    3, 1,  16, 4,  tid);
  load_bias(W + OFF_G3BIH, p.g3bih + d * 3, 3, 16, tid);
  load_bias(W + OFF_G3BHH, p.g3bhh + d * 3, 3, 16, tid);
  __syncthreads();

  const int r16  = lane & 15;
  const int half = lane >> 4;

  const int ntiles = BATCH / TROWS;                      // 2048 per d
  for (int t = blockIdx.y * NWAVE + wave; t < ntiles; t += GRIDY * NWAVE) {
    const int base = t * TROWS;
    const size_t row_id = (size_t)(base + r16) * DSRC + d;

    // ---- stage 1: A <- [del_x_hat, 0, 0]; hQ <- Q ----
    if (lane < 16) {
      const float2 xy = *(const float2*)(p.del_x_hat + row_id * 2);
      A[r16 * LDA + 0] = xy.x; A[r16 * LDA + 1] = xy.y;
      A[r16 * LDA + 2] = 0.f;  A[r16 * LDA + 3] = 0.f;
    }
    {
      const float2 q2 = *(const float2*)(p.Q + row_id * 4 + half * 2);
      hQ[r16 * 4 + half * 2 + 0] = q2.x;
      hQ[r16 * 4 + half * 2 + 1] = q2.y;
    }
    __syncthreads();
    layer_gemm<4, 48, LDA, 4, LDA>(A, W + OFF_FC1W, Bf, lane);     // fc1
    __syncthreads();
    if (lane < 16) {            // fused bias+relu+l2norm (in1), row in registers
      float v[40]; float s = 0.f;
#pragma unroll
      for (int c = 0; c < 40; ++c) {
        float x = Bf[r16 * LDA + c] + W[OFF_FC1B + c];
        x = fmaxf(x, 0.f); v[c] = x; s += x * x;
      }
      float inv = 1.f / fmaxf(sqrtf(s), 1e-6f);
#pragma unroll
      for (int c = 0; c < 40; ++c) Bf[r16 * LDA + c] = v[c] * inv;
    }
    __syncthreads();

    // ---- gru1: Qn ----
    layer_gemm<40, 16, LDA, 44, LDA>(Bf, W + OFF_G1WIH, A,      lane);   // GI
    layer_gemm<4,  16, 4,   4,  LDA>(hQ, W + OFF_G1WHH, A + 16, lane);   // GH
    __syncthreads();
    gru_combine<4>(A, A + 16, W + OFF_G1BIH, W + OFF_G1BHH, hQ, lane);
    __syncthreads();

    // ---- fc2(del_x_til); in2 = l2norm([Qn, relu(fc2)]);  hS <- Sigma ----
    if (lane < 16) {
      const float2 xy = *(const float2*)(p.del_x_til + row_id * 2);
      A[r16 * LDA + 0] = xy.x; A[r16 * LDA + 1] = xy.y;
      A[r16 * LDA + 2] = 0.f;  A[r16 * LDA + 3] = 0.f;
    }
    {
      const float2 s2 = *(const float2*)(p.Sigma + row_id * 4 + half * 2);
      hS[r16 * 4 + half * 2 + 0] = s2.x;
      hS[r16 * 4 + half * 2 + 1] = s2.y;
    }
    __syncthreads();
    layer_gemm<4, 48, LDA, 4, LDA>(A, W + OFF_FC2W, Bf, lane);     // fc2
    __syncthreads();
    if (lane < 16) {            // fused concat + bias+relu + l2norm -> A[0..43]
      float v[44]; float s = 0.f;
#pragma unroll
      for (int c = 0; c < 4; ++c) { float x = hQ[r16 * 4 + c]; v[c] = x; s += x * x; }
#pragma unroll
      for (int c = 0; c < 40; ++c) {
        float x = Bf[r16 * LDA + c] + W[OFF_FC2B + c];
        x = fmaxf(x, 0.f); v[4 + c] = x; s += x * x;
      }
      float inv = 1.f / fmaxf(sqrtf(s), 1e-6f);
#pragma unroll
      for (int c = 0; c < 44; ++c) A[r16 * LDA + c] = v[c] * inv;
    }
    __syncthreads();

    // ---- gru2: Sigman ----
    layer_gemm<44, 16, LDA, 44, LDA>(A,  W + OFF_G2WIH, Bf,      lane);  // GI
    layer_gemm<4,  16, 4,   4,  LDA>(hS, W + OFF_G2WHH, Bf + 16, lane);  // GH
    __syncthreads();
    gru_combine<4>(Bf, Bf + 16, W + OFF_G2BIH, W + OFF_G2BHH, hS, lane);
    __syncthreads();

    // ---- fc3(Sigman), fc4([dy_til,dy]); in3 = l2norm(cat);  sn <- S ----
    if (lane < 16) {
      A[r16 * LDA + 0] = p.del_y_til[row_id];
      A[r16 * LDA + 1] = p.del_y[row_id];
      A[r16 * LDA + 2] = 0.f; A[r16 * LDA + 3] = 0.f;
      sn[r16 * 4 + 0] = p.S[row_id];
      sn[r16 * 4 + 1] = 0.f; sn[r16 * 4 + 2] = 0.f; sn[r16 * 4 + 3] = 0.f;
    }
    __syncthreads();
    layer_gemm<4, 16, 4,   4, LDA>(hS, W + OFF_FC3W, Bf,      lane);     // fc3 -> Bf col 0
    layer_gemm<4, 48, LDA, 4, LDA>(A,  W + OFF_FC4W, Bf + 16, lane);     // fc4 -> cols 16..55
    __syncthreads();
    if (lane < 16) {            // fused concat + bias+relu + l2norm(41) -> A[0..43]
      float v[41]; float s = 0.f;
      {
        float x = Bf[r16 * LDA + 0] + W[OFF_FC3B + 0];
        x = fmaxf(x, 0.f); v[0] = x; s += x * x;
      }
#pragma unroll
      for (int c = 0; c < 40; ++c) {
        float x = Bf[r16 * LDA + 16 + c] + W[OFF_FC4B + c];
        x = fmaxf(x, 0.f); v[1 + c] = x; s += x * x;
      }
      float inv = 1.f / fmaxf(sqrtf(s), 1e-6f);
#pragma unroll
      for (int c = 0; c < 41; ++c) A[r16 * LDA + c] = v[c] * inv;
      A[r16 * LDA + 41] = 0.f; A[r16 * LDA + 42] = 0.f; A[r16 * LDA + 43] = 0.f;
    }
    __syncthreads();

    // ---- gru3: Sn ----
    layer_gemm<44, 16, LDA, 44, LDA>(A,  W + OFF_G3WIH, Bf,      lane);  // GI
    layer_gemm<4,  16, 4,   4,  LDA>(sn, W + OFF_G3WHH, Bf + 16, lane);  // GH
    __syncthreads();
    gru_combine<1>(Bf, Bf + 16, W + OFF_G3BIH, W + OFF_G3BHH, sn, lane);
    __syncthreads();

    // ---- K = fc5b(relu(fc5a([Sigman, Sn]))) ----
    if (lane < 16) {
      A[r16 * LDA + 0] = hS[r16 * 4 + 0];
      A[r16 * LDA + 1] = hS[r16 * 4 + 1];
      A[r16 * LDA + 2] = hS[r16 * 4 + 2];
      A[r16 * LDA + 3] = hS[r16 * 4 + 3];
      A[r16 * LDA + 4] = sn[r16 * 4 + 0];
      A[r16 * LDA + 5] = 0.f; A[r16 * LDA + 6] = 0.f; A[r16 * LDA + 7] = 0.f;
    }
    __syncthreads();
    layer_gemm<8, 112, LDA, 12, LDA>(A, W + OFF_FC5AW, Bf, lane);
    __syncthreads();
    bias_relu_half<100>(Bf, W + OFF_FC5AB, lane);
    __syncthreads();
    layer_gemm<100, 16, LDA, 100, LDA>(Bf, W + OFF_FC5BW, A, lane);      // K -> A cols 0..1
    __syncthreads();
    if (lane < 16) {                                                     // + bias (no relu)
      A[r16 * LDA + 0] += W[OFF_FC5BB + 0];
      A[r16 * LDA + 1] += W[OFF_FC5BB + 1];
      // fc6 input [Sn, K, 0] -> A cols 16..19
      A[r16 * LDA + 16] = sn[r16 * 4 + 0];
      A[r16 * LDA + 17] = A[r16 * LDA + 0];
      A[r16 * LDA + 18] = A[r16 * LDA + 1];
      A[r16 * LDA + 19] = 0.f;
    }
    __syncthreads();

    // ---- fc6 -> Bf cols 0..3 ----
    layer_gemm<4, 16, LDA, 4, LDA>(A + 16, W + OFF_FC6W, Bf, lane);
    __syncthreads();
    if (lane < 16) {   // bias+relu on fc6, then build fc7 input [Sigman, fc6out] -> Bf cols 16..23
#pragma unroll
      for (int c = 0; c < 4; ++c) {
        float x = Bf[r16 * LDA + c] + W[OFF_FC6B + c];
        x = fmaxf(x, 0.f);
        Bf[r16 * LDA + 16 + 4 + c] = x;
        Bf[r16 * LDA + 16 + c] = hS[r16 * 4 + c];
      }
    }
    __syncthreads();

    // ---- Sigma_next = relu(fc7) -> A cols 16..19 ----
    layer_gemm<8, 16, LDA, 12, LDA>(Bf + 16, W + OFF_FC7W, A + 16, lane);
    __syncthreads();
    if (lane < 16) {
#pragma unroll
      for (int c = 0; c < 4; ++c) {
        float x = A[r16 * LDA + 16 + c] + W[OFF_FC7B + c];
        A[r16 * LDA + 16 + c] = fmaxf(x, 0.f);
      }
      // ---- output row: [K(2), Qn(4), Sigma_next(4), Sn(1)] ----
      float* o = p.out + row_id * 11;
      o[0]  = A[r16 * LDA + 0];
      o[1]  = A[r16 * LDA + 1];
      o[2]  = hQ[r16 * 4 + 0];
      o[3]  = hQ[r16 * 4 + 1];
      o[4]  = hQ[r16 * 4 + 2];
      o[5]  = hQ[r16 * 4 + 3];
      o[6]  = A[r16 * LDA + 16 + 0];
      o[7]  = A[r16 * LDA + 16 + 1];
      o[8]  = A[r16 * LDA + 16 + 2];
      o[9]  = A[r16 * LDA + 16 + 3];
      o[10] = sn[r16 * 4 + 0];
    }
    __syncthreads();   // buffers reused next iteration
  }
}

extern "C" void kernel_launch(void* const* d_in, const int* in_sizes, int n_in,
                              void* d_out, int out_size, void* d_ws, size_t ws_size,
                              hipStream_t stream) {
  (void)in_sizes; (void)n_in; (void)out_size; (void)d_ws; (void)ws_size;
  KNParams p;
  p.del_y_til = (const float*)d_in[0];
  p.del_y     = (const float*)d_in[1];
  p.del_x_til = (const float*)d_in[2];
  p.del_x_hat = (const float*)d_in[3];
  p.Q         = (const float*)d_in[4];
  p.Sigma     = (const float*)d_in[5];
  p.S         = (const float*)d_in[6];
  p.fc1_w  = (const float*)d_in[7];   p.fc1_b  = (const float*)d_in[8];
  p.fc2_w  = (const float*)d_in[9];   p.fc2_b  = (const float*)d_in[10];
  p.fc3_w  = (const float*)d_in[11];  p.fc3_b  = (const float*)d_in[12];
  p.fc4_w  = (const float*)d_in[13];  p.fc4_b  = (const float*)d_in[14];
  p.fc5a_w = (const float*)d_in[15];  p.fc5a_b = (const float*)d_in[16];
  p.fc5b_w = (const float*)d_in[17];  p.fc5b_b = (const float*)d_in[18];
  p.fc6_w  = (const float*)d_in[19];  p.fc6_b  = (const float*)d_in[20];
  p.fc7_w  = (const float*)d_in[21];  p.fc7_b  = (const float*)d_in[22];
  p.g1wih = (const float*)d_in[23]; p.g1whh = (const float*)d_in[24];
  p.g1bih = (const float*)d_in[25]; p.g1bhh = (const float*)d_in[26];
  p.g2wih = (const float*)d_in[27]; p.g2whh = (const float*)d_in[28];
  p.g2bih = (const float*)d_in[29]; p.g2bhh = (const float*)d_in[30];
  p.g3wih = (const float*)d_in[31]; p.g3whh = (const float*)d_in[32];
  p.g3bih = (const float*)d_in[33]; p.g3bhh = (const float*)d_in[34];
  p.out = (float*)d_out;

  dim3 grid(DSRC, GRIDY, 1);
  dim3 block(NWAVE * 32, 1, 1);
  kalmannet_wmma_kernel<<<grid, block, 0, stream>>>(p);
}